// repa_conv_layer_22565758173777
// MI455X (gfx1250) — compile-verified
//
#include <hip/hip_runtime.h>

typedef __attribute__((ext_vector_type(2))) float v2f;
typedef __attribute__((ext_vector_type(4))) float v4f;
typedef __attribute__((ext_vector_type(8))) float v8f;

#define N_NODES   163842
#define IN_FEATS  32
#define OUT_FEATS 64
#define TAPS      25
#define K_TOTAL   800          // 25 * 32
#define M_TILE    32           // nodes per workgroup (2 M-tiles of 16)
#define H_STRIDE  804          // 804 % 64 = 36 -> conflict-free A-operand ds_load_b64
#define KC        80           // K-chunk of W staged in LDS (10 chunks)
#define NCHUNK    (K_TOTAL / KC)
#define WT_STRIDE 72           // rows k and k+2 land in disjoint LDS bank halves
#define THREADS   256          // 8 wave32

#if __has_builtin(__builtin_amdgcn_global_load_async_to_lds_b32) && \
    __has_builtin(__builtin_amdgcn_s_wait_asynccnt)
#define USE_ASYNC_LDS 1
#else
#define USE_ASYNC_LDS 0
#endif

typedef __attribute__((address_space(1))) int g_int;
typedef __attribute__((address_space(3))) int l_int;

__device__ __forceinline__ void stage_w_chunk(const float* __restrict__ W,
                                              float* __restrict__ buf,
                                              int kc, int tid)
{
    // transpose-stage W[:, kc:kc+KC] -> buf[kk][n]   (5120 elems, 20 per thread)
    for (int i = tid; i < OUT_FEATS * KC; i += THREADS) {
        const int n  = i / KC;
        const int kk = i % KC;
#if USE_ASYNC_LDS
        // CDNA5 async DMA: global -> LDS, no VGPR round-trip, tracked by ASYNCcnt
        __builtin_amdgcn_global_load_async_to_lds_b32(
            (g_int*)(W + n * K_TOTAL + kc + kk),
            (l_int*)(buf + kk * WT_STRIDE + n),
            0, 0);
#else
        buf[kk * WT_STRIDE + n] = W[n * K_TOTAL + kc + kk];
#endif
    }
}

__device__ __forceinline__ void wait_w_chunk()
{
#if USE_ASYNC_LDS
    __builtin_amdgcn_s_wait_asynccnt(0);
#endif
    __syncthreads();
}

__global__ __launch_bounds__(THREADS) void repa_conv_wmma_f32(
    const float* __restrict__ x,
    const float* __restrict__ nw,
    const float* __restrict__ W,
    const float* __restrict__ bias,
    const int*   __restrict__ nidx,
    float* __restrict__ out)
{
    __shared__ __align__(16) float h[M_TILE * H_STRIDE];       // 102.9 KB
    __shared__ __align__(16) float wt[2][KC * WT_STRIDE];      // 2 x 23.0 KB (double buffer)

    const int tid  = threadIdx.x;
    const int lane = tid & 31;
    const int wave = tid >> 5;
    const int m0g  = blockIdx.x * M_TILE;

    // ---------------- Phase 1: gather + 3-point weighted sum -> h[m][k] ----------------
    for (int p = tid; p < M_TILE * TAPS; p += THREADS) {
        const int m    = p / TAPS;
        const int g    = p % TAPS;
        const int node = m0g + m;
        const bool valid = node < N_NODES;
        int   i0 = 0, i1 = 0, i2 = 0;
        float w0 = 0.f, w1 = 0.f, w2 = 0.f;
        if (valid) {
            const int base = node * 75 + g * 3;
            i0 = nidx[base + 0] - 1;          // 1-based -> 0-based
            i1 = nidx[base + 1] - 1;
            i2 = nidx[base + 2] - 1;
            w0 = nw[base + 0]; w1 = nw[base + 1]; w2 = nw[base + 2];
        }
        const v4f* x0 = (const v4f*)(x + (long)i0 * IN_FEATS);
        const v4f* x1 = (const v4f*)(x + (long)i1 * IN_FEATS);
        const v4f* x2 = (const v4f*)(x + (long)i2 * IN_FEATS);
        float* hrow = h + m * H_STRIDE + g * IN_FEATS;
#pragma unroll
        for (int c4 = 0; c4 < IN_FEATS / 4; ++c4) {
            v4f v = x0[c4] * w0 + x1[c4] * w1 + x2[c4] * w2;
            *(v4f*)(hrow + c4 * 4) = v;       // ds_store_b128, 16B aligned
        }
    }
    __syncthreads();

    // ---------------- Phase 2: (32 x 800) @ (800 x 64) via V_WMMA_F32_16X16X4_F32 ------
    // wave -> output tile: rows [m0g + 16*mt, +16), cols [16*(wave&3), +16)
    const int mt  = wave >> 2;
    const int n0  = (wave & 3) * 16;
    const int l16 = lane & 15;
    const int hi  = lane >> 4;                // 0: lanes 0-15, 1: lanes 16-31

    // A operand: VGPR0 = h[row][k + 2*hi], VGPR1 = h[row][k + 2*hi + 1] (one ds_load_b64)
    const float* hA = h + (mt * 16 + l16) * H_STRIDE + hi * 2;

    v8f acc = {};                             // fp32 C/D, 800-deep fp32 accumulation

    // prologue: stage chunk 0
    stage_w_chunk(W, wt[0], 0, tid);
    wait_w_chunk();

    for (int c = 0; c < NCHUNK; ++c) {
        const float* cur = wt[c & 1];
        // kick off DMA of the next chunk; it runs under the WMMA stream below
        if (c + 1 < NCHUNK)
            stage_w_chunk(W, wt[(c + 1) & 1], (c + 1) * KC, tid);

        // B operand: VGPR0 = cur[kk + 2*hi][n0+l16], VGPR1 = cur[kk + 2*hi + 1][n0+l16]
        const float* wB = cur + hi * 2 * WT_STRIDE + n0 + l16;
#pragma unroll
        for (int kk = 0; kk < KC; kk += 4) {
            v2f a = *(const v2f*)(hA + c * KC + kk);
            v2f b;
            b.x = wB[(kk + 0) * WT_STRIDE];
            b.y = wB[(kk + 1) * WT_STRIDE];
            acc = __builtin_amdgcn_wmma_f32_16x16x4_f32(
                /*neg_a=*/false, a, /*neg_b=*/false, b,
                /*c_mod=*/(short)0, acc, /*reuse_a=*/false, /*reuse_b=*/false);
        }
        wait_w_chunk();   // s_wait_asynccnt 0 (+ barrier): next buffer ready, cur reusable
    }

    // ---------------- Epilogue: bias + store (C/D layout: VGPR v -> M = v + 8*hi) ------
    const float bv = bias[n0 + l16];
#pragma unroll
    for (int v = 0; v < 8; ++v) {
        const int node = m0g + mt * 16 + v + hi * 8;
        if (node < N_NODES) {
            out[(long)node * OUT_FEATS + n0 + l16] = acc[v] + bv;
        }
    }
}

extern "C" void kernel_launch(void* const* d_in, const int* in_sizes, int n_in,
                              void* d_out, int out_size, void* d_ws, size_t ws_size,
                              hipStream_t stream) {
    const float* x   = (const float*)d_in[0];
    const float* nw  = (const float*)d_in[1];
    const float* W   = (const float*)d_in[2];
    const float* b   = (const float*)d_in[3];
    const int*   idx = (const int*)d_in[4];
    float* out = (float*)d_out;

    const int grid = (N_NODES + M_TILE - 1) / M_TILE;   // 5121 workgroups, 8 wave32 each
    repa_conv_wmma_f32<<<grid, THREADS, 0, stream>>>(x, nw, W, b, idx, out);
}